// TimestepNorm_25563645345977
// MI455X (gfx1250) — compile-verified
//
#include <hip/hip_runtime.h>

// TimestepNorm for MI455X (gfx1250, wave32).
// One wave per (batch, group) chain: 256 blocks x 32 threads.
// Per 16-timestep tile:
//   - 16x global_load_b64 (non-temporal) in WMMA A-layout, double-buffered in VGPRs
//   - 16x v_wmma_f32_16x16x4_f32 against B = 1/64  => per-timestep group means,
//     broadcast across all lanes by construction (all D columns identical)
//   - 1x ds_swizzle SWAPX16 per acc reg to merge row halves
//   - Welford scan compiles to the SALU path (s_cvt/v_s_rcp_f32), co-executing
//     with WMMA; per-lane stat select via v_cndmask
//   - normalize from the same registers, non-temporal global_store_b64
// x read once (256MB), y written once (256MB): ~22us roofline at 23.3 TB/s.
// global_prefetch_b8 four tiles ahead (WGP locality) covers HBM latency at the
// low (256-wave) occupancy the sequential scan allows; NT hints keep the
// write-once y stream from displacing prefetched x lines in L2.

typedef __attribute__((ext_vector_type(2))) float v2f;
typedef __attribute__((ext_vector_type(8))) float v8f;

#define B_  8
#define L_  4096
#define D_  2048
#define G_  32
#define GS_ 64
#define NT  (L_ / 16)      // 256 tiles of 16 timesteps
#define PF  4              // prefetch distance in tiles
#define EPSV 1e-5f

__device__ __forceinline__ float lane_swap16(float x) {
  // ds_swizzle_b32 group-of-32: xor=0x10, or=0, and=0x1f  => SWAPX16
  return __int_as_float(__builtin_amdgcn_ds_swizzle(__float_as_int(x), 0x401F));
}

__global__ __launch_bounds__(32) void TimestepNorm_kernel(
    const float* __restrict__ x, const unsigned char* __restrict__ mask,
    const int* __restrict__ prev_count, const float* __restrict__ prev_mean,
    const float* __restrict__ prev_var, const float* __restrict__ weight,
    const float* __restrict__ bias, float* __restrict__ out) {
  const int chain = blockIdx.x;          // 0..255
  const int b = chain / G_;
  const int g = chain % G_;
  const int l  = threadIdx.x;            // 0..31
  const int hl = l >> 4;                 // half-wave id
  const int tl = l & 15;                 // row (timestep-in-tile) this lane owns

  const float* xbase = x   + (size_t)b * L_ * D_ + (size_t)g * GS_;
  float*       ybase = out + (size_t)b * L_ * D_ + (size_t)g * GS_;
  const unsigned char* mbase = mask + (size_t)b * L_;

  // Per-lane weight/bias slice: element k = 4c + 2*hl + {0,1}, c = 0..15
  v2f wv[16], bv[16];
  {
    const float* wb = weight + g * GS_ + 2 * hl;
    const float* bb = bias   + g * GS_ + 2 * hl;
#pragma unroll
    for (int c = 0; c < 16; ++c) {
      wv[c] = *(const v2f*)(wb + 4 * c);
      bv[c] = *(const v2f*)(bb + 4 * c);
    }
  }

  int   cnt  = prev_count[b];
  float mean = prev_mean[chain];
  float var  = prev_var[chain];

  const v2f bmat = {1.0f / GS_, 1.0f / GS_};   // ones/GS, WMMA B operand

  // WMMA A-layout tile load: lane (hl,tl) -> row tl, elems 4c + 2*hl + {0,1}.
  // Non-temporal: each 256B row is consumed by back-to-back loads, then never
  // re-read from memory (normalize reuses the registers).
  auto load_tile = [&](int j, v2f (&buf)[16]) __attribute__((always_inline)) {
    const float* row = xbase + (size_t)(j * 16 + tl) * D_ + 2 * hl;
#pragma unroll
    for (int c = 0; c < 16; ++c)
      buf[c] = __builtin_nontemporal_load((const v2f*)(row + 4 * c));
  };

  auto prefetch_tile = [&](int j) __attribute__((always_inline)) {
    // lanes 0-15 prefetch row starts, lanes 16-31 hit +128B of the 256B row;
    // locality 3 => nearest-scope prefetch (pull into WGP-level caches)
    __builtin_prefetch(xbase + (size_t)(j * 16 + tl) * D_ + 32 * hl, 0, 3);
  };

  auto process = [&](int j, v2f (&cur)[16]) __attribute__((always_inline)) {
    // --- per-timestep group means via 16 chained f32 WMMAs ------------------
    v8f acc = {};
#pragma unroll
    for (int c = 0; c < 16; ++c) {
      acc = __builtin_amdgcn_wmma_f32_16x16x4_f32(
          /*neg_a=*/false, cur[c], /*neg_b=*/false, bmat,
          /*c_mod=*/(short)0, acc, /*reuse_a=*/false, /*reuse_b=*/false);
    }
    // acc[r] (lanes 0-15) = m[r]; (lanes 16-31) = m[r+8]; merge both halves
    float marr[16];
#pragma unroll
    for (int r = 0; r < 8; ++r) {
      float sw = lane_swap16(acc[r]);
      marr[r]     = (hl == 0) ? acc[r] : sw;
      marr[r + 8] = (hl == 0) ? sw     : acc[r];
    }

    // --- Welford scan over the 16 timesteps (scalarizes to SALU) ------------
    const unsigned int* mq = (const unsigned int*)(mbase + j * 16);
    unsigned int mw0 = mq[0], mw1 = mq[1], mw2 = mq[2], mw3 = mq[3];
    float selM = 0.0f, selR = 0.0f;
#pragma unroll
    for (int t = 0; t < 16; ++t) {
      unsigned int mw = (t < 4) ? mw0 : (t < 8) ? mw1 : (t < 12) ? mw2 : mw3;
      int mv = (int)((mw >> (8 * (t & 3))) & 0xFFu);
      float valid = mv ? 1.0f : 0.0f;
      float cprev_safe = (float)(cnt > 1 ? cnt : 1);
      cnt += (mv ? 1 : 0);
      float csafe = (float)(cnt > 1 ? cnt : 1);
      float rc = __builtin_amdgcn_rcpf(csafe);
      float delta = marr[t] - mean;
      float mean_new = mean + delta * valid * rc;
      float m2v = var * cprev_safe + delta * (marr[t] - mean_new) * valid;
      var = m2v * rc;
      mean = mean_new;
      float rstd = __builtin_amdgcn_rsqf(var + EPSV);
      // lane picks the stats of the row it normalizes (v_cndmask, no LDS)
      selM = (tl == t) ? mean : selM;
      selR = (tl == t) ? rstd : selR;
    }

    // --- normalize the same registers; NT stores (write-once stream) -------
    float* yrow = ybase + (size_t)(j * 16 + tl) * D_ + 2 * hl;
#pragma unroll
    for (int c = 0; c < 16; ++c) {
      v2f xv = cur[c];
      v2f yv;
      yv.x = (xv.x - selM) * selR * wv[c].x + bv[c].x;
      yv.y = (xv.y - selM) * selR * wv[c].y + bv[c].y;
      __builtin_nontemporal_store(yv, (v2f*)(yrow + 4 * c));
    }
  };

  // --- main loop: double-buffered register tiles, prefetch PF tiles ahead ---
  v2f bufA[16], bufB[16];
  load_tile(0, bufA);
  load_tile(1, bufB);
  for (int j = 0; j < NT; j += 2) {
    if (j + PF     < NT) prefetch_tile(j + PF);
    if (j + PF + 1 < NT) prefetch_tile(j + PF + 1);
    process(j, bufA);
    if (j + 2 < NT) load_tile(j + 2, bufA);
    process(j + 1, bufB);
    if (j + 3 < NT) load_tile(j + 3, bufB);
  }

  // --- final stats ----------------------------------------------------------
  const size_t YN = (size_t)B_ * L_ * D_;
  if (l == 0) {
    float* out_mean = out + YN + 8;            // after count slots
    float* out_var  = out + YN + 8 + B_ * G_;
    out_mean[chain] = mean;
    out_var[chain]  = var;
    if (g == 0) out[YN + b] = (float)cnt;      // final count per batch
  }
}

extern "C" void kernel_launch(void* const* d_in, const int* in_sizes, int n_in,
                              void* d_out, int out_size, void* d_ws, size_t ws_size,
                              hipStream_t stream) {
  (void)in_sizes; (void)n_in; (void)d_ws; (void)ws_size; (void)out_size;
  const float*         x    = (const float*)d_in[0];
  const unsigned char* pm   = (const unsigned char*)d_in[1];
  const int*           pcnt = (const int*)d_in[2];
  const float*         pmu  = (const float*)d_in[3];
  const float*         pvar = (const float*)d_in[4];
  const float*         w    = (const float*)d_in[5];
  const float*         bia  = (const float*)d_in[6];
  float*               out  = (float*)d_out;

  dim3 grid(B_ * G_);   // 256 chains, one wave32 each
  dim3 block(32);
  TimestepNorm_kernel<<<grid, block, 0, stream>>>(x, pm, pcnt, pmu, pvar, w, bia, out);
}